// RNN_23106924052493
// MI455X (gfx1250) — compile-verified
//
#include <hip/hip_runtime.h>
#include <hip/hip_bf16.h>

// Problem dims (compile-time constants from the reference)
#define T_DIM 1024
#define B_DIM 64
#define F_DIM 256
#define H_DIM 512

typedef __attribute__((ext_vector_type(2))) float v2f;
typedef __attribute__((ext_vector_type(4))) float v4f;
typedef __attribute__((ext_vector_type(8))) float v8f;
typedef __attribute__((ext_vector_type(4))) int   v4i;

typedef __attribute__((address_space(1))) v4i as1_v4i;
typedef __attribute__((address_space(3))) v4i as3_v4i;

#if __has_builtin(__builtin_amdgcn_global_load_async_to_lds_b128)
#define HAVE_ASYNC_LDS 1
#else
#define HAVE_ASYNC_LDS 0
#endif

__device__ __forceinline__ void wait_asynccnt0() {
#if __has_builtin(__builtin_amdgcn_s_wait_asynccnt)
    __builtin_amdgcn_s_wait_asynccnt(0);
#else
    asm volatile("s_wait_asynccnt 0" ::: "memory");
#endif
}

// ---------------------------------------------------------------------------
// Kernel A: wx_out[m, h] = sum_f x[m, f] * Wx[h, f] + bx[h]
//   M = T*B = 65536, N = H = 512, K = F = 256.  One 16x16 tile per wave.
// ---------------------------------------------------------------------------
__global__ __launch_bounds__(256) void k_in_proj(const float* __restrict__ x,
                                                 const float* __restrict__ Wx,
                                                 const float* __restrict__ bx,
                                                 float* __restrict__ hbuf) {
    const int lane = threadIdx.x & 31;
    const int wave = threadIdx.x >> 5;
    const int tile = blockIdx.x * 8 + wave;
    const int ntn  = H_DIM / 16;              // 32 n-tiles
    const int mt   = tile / ntn;
    const int nt   = tile % ntn;

    const int m    = mt * 16 + (lane & 15);
    const int n    = nt * 16 + (lane & 15);
    const int koff = (lane >> 4) * 2;         // lanes 0-15 -> K{0,1}, 16-31 -> K{2,3}

    const float* arow = x  + (size_t)m * F_DIM + koff;
    const float* brow = Wx + (size_t)n * F_DIM + koff;

    v8f acc = {};
#pragma unroll 8
    for (int k = 0; k < F_DIM; k += 4) {
        v2f a = *(const v2f*)(arow + k);
        v2f b = *(const v2f*)(brow + k);
        acc = __builtin_amdgcn_wmma_f32_16x16x4_f32(false, a, false, b,
                                                    (short)0, acc, false, false);
    }

    const float bias = bx[n];
    float* out = hbuf + (size_t)(mt * 16 + (lane >> 4) * 8) * H_DIM + n;
#pragma unroll
    for (int r = 0; r < 8; ++r)
        out[(size_t)r * H_DIM] = acc[r] + bias;
}

// ---------------------------------------------------------------------------
// Kernel B: persistent recurrence  h_t = wx_out[t] + h_{t-1} @ U^T + bu
//   8 workgroups x 512 threads (16 waves). WG wg owns columns [wg*64, wg*64+64).
//   LDS: U slice (64x512) + h_{t-1} stage (64x512), both padded to stride 516
//   for conflict-free float2 reads. h_{t-1} staged per step via async
//   global->LDS copies (ASYNCcnt). Device-scope atomic barrier per step.
// ---------------------------------------------------------------------------
#define RWG 8
#define PAD_STRIDE 516   // 512 + 4 pad floats

__global__ __launch_bounds__(512) void k_recur(const float* __restrict__ U,
                                               const float* __restrict__ bu,
                                               float* __restrict__ hbuf,
                                               unsigned* __restrict__ ctr) {
    extern __shared__ float smem[];                   // 2 * 64*516 floats
    float* Ulds = smem;                               // U slice
    float* Hlds = smem + 64 * PAD_STRIDE;             // h_{t-1} stage

    const int lane = threadIdx.x & 31;
    const int wave = threadIdx.x >> 5;        // 0..15
    const int j0   = blockIdx.x * 64;         // this WG's H-column base

    // stage U slice once: rows j0..j0+63, all 512 K values
    for (int i = threadIdx.x; i < 64 * H_DIM; i += 512) {
        const int row = i >> 9;               // /512
        const int col = i & (H_DIM - 1);
        Ulds[row * PAD_STRIDE + col] = U[(size_t)(j0 + row) * H_DIM + col];
    }
    __syncthreads();

    const int mbase = (wave & 3) * 16;                  // 4 m-tiles cover B=64
    const int m     = mbase + (lane & 15);              // batch row 0..63
    const int nl    = (wave >> 2) * 16 + (lane & 15);   // local column 0..63
    const int n     = j0 + nl;
    const int koff  = (lane >> 4) * 2;
    const float bias = bu[n];
    const float* ul = &Ulds[nl * PAD_STRIDE + koff];
    const float* al = &Hlds[m  * PAD_STRIDE + koff];

    for (int t = 0; t < T_DIM; ++t) {
        float* hslab = hbuf + (size_t)t * (B_DIM * H_DIM);

        v8f acc = {};
        if (t > 0) {
            // ---- stage h_{t-1} (64 x 512 fp32 = 128 KB) into LDS ----
            const float* hprev = hbuf + (size_t)(t - 1) * (B_DIM * H_DIM);
            for (int i = threadIdx.x; i < (B_DIM * H_DIM) / 4; i += 512) {
                const int row = i >> 7;       // 128 float4 groups per row
                const int grp = i & 127;
                const float* gsrc = hprev + (size_t)row * H_DIM + grp * 4;
                float*       ldst = Hlds + row * PAD_STRIDE + grp * 4;
#if HAVE_ASYNC_LDS
                __builtin_amdgcn_global_load_async_to_lds_b128(
                    (as1_v4i*)gsrc, (as3_v4i*)ldst, 0, 0);
#else
                *(v4f*)ldst = *(const v4f*)gsrc;
#endif
            }
#if HAVE_ASYNC_LDS
            wait_asynccnt0();
#endif
            __syncthreads();

            // ---- 16x16 tile GEMM from LDS: A = h_{t-1}, B = U^T slice ----
#pragma unroll 8
            for (int k = 0; k < H_DIM; k += 4) {
                v2f a = *(const v2f*)(al + k);      // h_{t-1} row m (LDS)
                v2f b = *(const v2f*)(ul + k);      // U row n (LDS)
                acc = __builtin_amdgcn_wmma_f32_16x16x4_f32(false, a, false, b,
                                                            (short)0, acc, false, false);
            }
        }

        // h_t = wx_out[t] (already in hslab) + acc + bu ; overwrite in place
        float* cell = hslab + (size_t)(mbase + (lane >> 4) * 8) * H_DIM + n;
#pragma unroll
        for (int r = 0; r < 8; ++r) {
            cell[(size_t)r * H_DIM] = cell[(size_t)r * H_DIM] + acc[r] + bias;
        }

        // device-scope split barrier between steps
        __builtin_amdgcn_fence(__ATOMIC_RELEASE, "agent");   // flush h_t
        __syncthreads();
        if (threadIdx.x == 0) {
            atomicAdd(ctr, 1u);
            const unsigned goal = (unsigned)(RWG * (t + 1));
            while (__hip_atomic_load(ctr, __ATOMIC_RELAXED,
                                     __HIP_MEMORY_SCOPE_AGENT) < goal) {
                __builtin_amdgcn_s_sleep(1);
            }
        }
        __syncthreads();
        __builtin_amdgcn_fence(__ATOMIC_ACQUIRE, "agent");   // invalidate caches
    }
}

// ---------------------------------------------------------------------------
// Kernel C: out[m, f] = tanh( sum_h hs[m, h] * Wo[f, h] + bo[f] )
//   M = 65536, N = F = 256, K = H = 512.
// ---------------------------------------------------------------------------
__global__ __launch_bounds__(256) void k_out_proj(const float* __restrict__ hs,
                                                  const float* __restrict__ Wo,
                                                  const float* __restrict__ bo,
                                                  float* __restrict__ out) {
    const int lane = threadIdx.x & 31;
    const int wave = threadIdx.x >> 5;
    const int tile = blockIdx.x * 8 + wave;
    const int ntn  = F_DIM / 16;              // 16 n-tiles
    const int mt   = tile / ntn;
    const int nt   = tile % ntn;

    const int m    = mt * 16 + (lane & 15);
    const int n    = nt * 16 + (lane & 15);
    const int koff = (lane >> 4) * 2;

    const float* arow = hs + (size_t)m * H_DIM + koff;
    const float* brow = Wo + (size_t)n * H_DIM + koff;

    v8f acc = {};
#pragma unroll 8
    for (int k = 0; k < H_DIM; k += 4) {
        v2f a = *(const v2f*)(arow + k);
        v2f b = *(const v2f*)(brow + k);
        acc = __builtin_amdgcn_wmma_f32_16x16x4_f32(false, a, false, b,
                                                    (short)0, acc, false, false);
    }

    const float bias = bo[n];
    float* o = out + (size_t)(mt * 16 + (lane >> 4) * 8) * F_DIM + n;
#pragma unroll
    for (int r = 0; r < 8; ++r)
        o[(size_t)r * F_DIM] = tanhf(acc[r] + bias);
}

// ---------------------------------------------------------------------------
extern "C" void kernel_launch(void* const* d_in, const int* in_sizes, int n_in,
                              void* d_out, int out_size, void* d_ws, size_t ws_size,
                              hipStream_t stream) {
    const float* x  = (const float*)d_in[0];   // [T,B,F]
    const float* Wx = (const float*)d_in[1];   // [H,F]
    const float* bx = (const float*)d_in[2];   // [H]
    const float* U  = (const float*)d_in[3];   // [H,H]
    const float* bu = (const float*)d_in[4];   // [H]
    const float* Wo = (const float*)d_in[5];   // [F,H]
    const float* bo = (const float*)d_in[6];   // [F]
    float* out = (float*)d_out;

    // workspace: [T*B*H] fp32 wx_out/hs buffer, then barrier counter
    float*    hbuf = (float*)d_ws;
    unsigned* ctr  = (unsigned*)((char*)d_ws + (size_t)T_DIM * B_DIM * H_DIM * sizeof(float));

    (void)hipMemsetAsync(ctr, 0, sizeof(unsigned), stream);

    // A: input projection. tiles = (65536/16)*(512/16) = 131072; 8 waves/block.
    const int blocksA = (T_DIM * B_DIM / 16) * (H_DIM / 16) / 8;   // 16384
    k_in_proj<<<blocksA, 256, 0, stream>>>(x, Wx, bx, hbuf);

    // B: persistent recurrence, 8 WGs x 512 threads, ~258 KB dynamic LDS each.
    const size_t ldsB = (size_t)2 * 64 * PAD_STRIDE * sizeof(float);
    k_recur<<<RWG, 512, ldsB, stream>>>(U, bu, hbuf, ctr);

    // C: output projection + tanh. tiles = 4096*16 = 65536; 8 waves/block.
    const int blocksC = (T_DIM * B_DIM / 16) * (F_DIM / 16) / 8;   // 8192
    k_out_proj<<<blocksC, 256, 0, stream>>>(hbuf, Wo, bo, out);
}